// SmokeCFDModel_38276748542150
// MI455X (gfx1250) — compile-verified
//
#include <hip/hip_runtime.h>
#include <hip/hip_bf16.h>

// CDNA5 / gfx1250 GNN pipeline. Memory-bound (≈7 GB traffic, 30 GFLOP), so all
// GEMMs run on V_WMMA_F32_16X16X4_F32 (exact fp32, matrix pipe), weight tiles
// are staged with double-buffered GLOBAL_LOAD_ASYNC_TO_LDS_B128 (ASYNCcnt),
// and the bandwidth-dominant edge scatter uses coalesced float4 reads, a
// global_prefetch lookahead, and native f32 global atomics.

typedef __attribute__((ext_vector_type(2))) float v2f;
typedef __attribute__((ext_vector_type(8))) float v8f;

__device__ __forceinline__ float gelu_exact(float x) {
    return 0.5f * x * (1.0f + erff(x * 0.70710678118654752f));
}

__device__ __forceinline__ v8f wmma_f32_k4(v2f a, v2f b, v8f c) {
    // D = A(16x4) * B(4x16) + C(16x16), all f32, wave32.
    return __builtin_amdgcn_wmma_f32_16x16x4_f32(false, a, false, b, (short)0, c,
                                                 false, false);
}

// Async global -> LDS copy (CDNA5 ASYNCcnt path). Each thread copies 4x16B.
// LDS byte address = low 32 bits of the flat pointer (workgroup-relative).
__device__ __forceinline__ void stage_W_async(const float* gsrc, float* lbuf,
                                              int tid) {
#pragma unroll
    for (int i = 0; i < 4; ++i) {
        unsigned lds = (unsigned)(uintptr_t)(lbuf + (i * 256 + tid) * 4);
        const float* g = gsrc + (i * 256 + tid) * 4;
        asm volatile("global_load_async_to_lds_b128 %0, %1, off"
                     :: "v"(lds), "v"(g)
                     : "memory");
    }
}

// ---------------------------------------------------------------- utilities
__global__ __launch_bounds__(256) void zero_kernel(float* __restrict__ p, long n4) {
    long i = (long)blockIdx.x * blockDim.x + threadIdx.x;
    long stride = (long)gridDim.x * blockDim.x;
    float4 z = {0.f, 0.f, 0.f, 0.f};
    for (; i < n4; i += stride) ((float4*)p)[i] = z;
}

__global__ __launch_bounds__(256) void degree_kernel(const int* __restrict__ coli,
                                                     float* __restrict__ deg, int E) {
    int e = blockIdx.x * blockDim.x + threadIdx.x;
    if (e < E) unsafeAtomicAdd(&deg[coli[e]], 1.0f);
}

__global__ __launch_bounds__(256) void invdeg_kernel(float* __restrict__ deg, int n) {
    int i = blockIdx.x * blockDim.x + threadIdx.x;
    if (i < n) deg[i] = 1.0f / fmaxf(deg[i], 1.0f);
}

// ------------------------------------------------------------- edge scatter
// thread t -> edge e = t/32, 16B chunk c = t%32 of the 512B feature row.
__global__ __launch_bounds__(256) void scatter_kernel(const float* __restrict__ h,
                                                      float* __restrict__ agg,
                                                      const int* __restrict__ rowi,
                                                      const int* __restrict__ coli,
                                                      long total) {
    long t = (long)blockIdx.x * blockDim.x + threadIdx.x;
    long stride = (long)gridDim.x * blockDim.x;
    for (; t < total; t += stride) {
        long e = t >> 5;
        int c = (int)(t & 31);
        // warm L2/WGP$ for a row we will gather ~2048 edges from now
        if (c == 0 && (e + 2048) * 32 < total) {
            __builtin_prefetch(h + (size_t)rowi[e + 2048] * 128, 0, 1);
        }
        int src = rowi[e];
        int dst = coli[e];
        float4 v = ((const float4*)(h + (size_t)src * 128))[c];
        float* d = agg + (size_t)dst * 128 + c * 4;
        unsafeAtomicAdd(d + 0, v.x);
        unsafeAtomicAdd(d + 1, v.y);
        unsafeAtomicAdd(d + 2, v.z);
        unsafeAtomicAdd(d + 3, v.w);
    }
}

// ------------------------------------------------------------- encoder GEMM
// h[r, :] = gelu(x[r, 0:16] @ W[16,128] + b), one wave per 16-row tile.
__global__ __launch_bounds__(256) void encoder_kernel(const float* __restrict__ x,
                                                      const float* __restrict__ W,
                                                      const float* __restrict__ bias,
                                                      float* __restrict__ h, int n) {
    int tid = threadIdx.x;
    int wave = tid >> 5, lane = tid & 31;
    int half = lane >> 4, r = lane & 15;
    int r0 = (blockIdx.x * 8 + wave) * 16;
    if (r0 >= n) return;  // wave-uniform exit; EXEC stays all-ones for WMMA
    int rowr = r0 + r;
    if (rowr >= n) rowr = n - 1;  // clamp loads; stores guarded below

    v8f acc[8] = {};
#pragma unroll
    for (int s = 0; s < 4; ++s) {
        int k = s * 4 + half * 2;
        v2f a;
        a.x = x[(size_t)rowr * 16 + k];
        a.y = x[(size_t)rowr * 16 + k + 1];
#pragma unroll
        for (int t = 0; t < 8; ++t) {
            int col = t * 16 + r;
            v2f b;
            b.x = W[k * 128 + col];
            b.y = W[(k + 1) * 128 + col];
            acc[t] = wmma_f32_k4(a, b, acc[t]);
        }
    }
#pragma unroll
    for (int t = 0; t < 8; ++t) {
        int col = t * 16 + r;
        float bc = bias[col];
#pragma unroll
        for (int g = 0; g < 8; ++g) {
            int row = r0 + g + half * 8;
            if (row < n) h[(size_t)row * 128 + col] = gelu_exact(acc[t][g] + bc);
        }
    }
}

// ---------------------------------------------------- fused tile GEMM kernel
// out[tile] = (RESIDUAL? h[tile] : 0) + gelu(Z @ W + bias)
//   CHUNKS=8: Z = [h | agg*invdeg]  (K=256, GNN layer)
//   CHUNKS=4: Z = h                 (K=128, decoder stage 1)
// K processed in 32-deep chunks. W chunks arrive via double-buffered async
// global->LDS copies (ASYNCcnt); per-wave 16x32 Z tiles staged in padded LDS.
template <int CHUNKS, bool RESIDUAL>
__global__ __launch_bounds__(256) void tile_gemm_kernel(
    const float* hsrc, const float* __restrict__ agg,
    const float* __restrict__ invdeg, const float* __restrict__ W,
    const float* __restrict__ bias, float* outp, int n) {
    __shared__ float sW[2][32 * 128];  // 2 x 16 KB double buffer
    __shared__ float sZ[8][16 * 36];   // 18 KB: per-wave 16x32 tile, stride 36

    int tid = threadIdx.x;
    int wave = tid >> 5, lane = tid & 31;
    int half = lane >> 4, r = lane & 15;
    int r0 = (blockIdx.x * 8 + wave) * 16;

    v8f acc[8] = {};

    // per-wave Z tile staging for one 32-deep chunk
    auto stage_Z = [&](int chunk) {
        bool isAgg = (CHUNKS == 8) && (chunk >= 4);
        int kbase = (chunk & 3) * 32;  // column window within the 128-wide src
        const float* src = isAgg ? agg : hsrc;
#pragma unroll
        for (int i = 0; i < 4; ++i) {
            int f = i * 32 + lane;        // float4 index in [0,128)
            int rl = f >> 3, c4 = f & 7;  // local row, float4 column
            int row = r0 + rl;
            int rowc = row < n ? row : n - 1;
            float4 v = ((const float4*)(src + (size_t)rowc * 128 + kbase))[c4];
            if (isAgg) {
                float sc = invdeg[rowc];
                v.x *= sc; v.y *= sc; v.z *= sc; v.w *= sc;
            }
            float* dst = &sZ[wave][rl * 36 + c4 * 4];
            dst[0] = v.x; dst[1] = v.y; dst[2] = v.z; dst[3] = v.w;
        }
    };
    // 8 WMMA K-steps over the staged chunk in sW[buf]
    auto compute = [&](int buf) {
        const float* wb = sW[buf];
#pragma unroll
        for (int s = 0; s < 8; ++s) {
            int kl = s * 4 + half * 2;
            v2f a;
            a.x = sZ[wave][r * 36 + kl];
            a.y = sZ[wave][r * 36 + kl + 1];
#pragma unroll
            for (int t = 0; t < 8; ++t) {
                int col = t * 16 + r;
                v2f b;
                b.x = wb[kl * 128 + col];
                b.y = wb[(kl + 1) * 128 + col];
                acc[t] = wmma_f32_k4(a, b, acc[t]);
            }
        }
    };

    stage_W_async(W, sW[0], tid);  // prologue: chunk 0 in flight
    int chunk = 0;
    for (; chunk < CHUNKS - 1; ++chunk) {
        stage_W_async(W + (size_t)(chunk + 1) * 32 * 128, sW[(chunk + 1) & 1], tid);
        stage_Z(chunk);
        // retire chunk's 4 async ops (next chunk's 4 may stay in flight)
        asm volatile("s_wait_asynccnt 4" ::: "memory");
        __syncthreads();
        compute(chunk & 1);
        __syncthreads();  // all waves done reading before buffer is re-filled
    }
    stage_Z(chunk);
    asm volatile("s_wait_asynccnt 0" ::: "memory");
    __syncthreads();
    compute(chunk & 1);

    // ---- epilogue: bias + gelu (+ residual), store
#pragma unroll
    for (int t = 0; t < 8; ++t) {
        int col = t * 16 + r;
        float bc = bias[col];
#pragma unroll
        for (int g = 0; g < 8; ++g) {
            int row = r0 + g + half * 8;
            if (row < n) {
                size_t off = (size_t)row * 128 + col;
                float v = gelu_exact(acc[t][g] + bc);
                outp[off] = RESIDUAL ? (hsrc[off] + v) : v;
            }
        }
    }
}

// ------------------------------------------------------- decoder final layer
__global__ __launch_bounds__(256) void dec2_kernel(const float* __restrict__ g1,
                                                   const float* __restrict__ W2,
                                                   const float* __restrict__ b2,
                                                   float* __restrict__ out, int n) {
    int i = blockIdx.x * blockDim.x + threadIdx.x;
    if (i >= n) return;
    float a0 = b2[0], a1 = b2[1], a2 = b2[2], a3 = b2[3];
    const float4* row = (const float4*)(g1 + (size_t)i * 128);
#pragma unroll 4
    for (int k4 = 0; k4 < 32; ++k4) {
        float4 v = row[k4];
        const float* w = W2 + k4 * 16;
        a0 += v.x * w[0]  + v.y * w[4]  + v.z * w[8]  + v.w * w[12];
        a1 += v.x * w[1]  + v.y * w[5]  + v.z * w[9]  + v.w * w[13];
        a2 += v.x * w[2]  + v.y * w[6]  + v.z * w[10] + v.w * w[14];
        a3 += v.x * w[3]  + v.y * w[7]  + v.z * w[11] + v.w * w[15];
    }
    float4 o = {a0, a1, a2, a3};
    ((float4*)out)[i] = o;
}

extern "C" void kernel_launch(void* const* d_in, const int* in_sizes, int n_in,
                              void* d_out, int out_size, void* d_ws, size_t ws_size,
                              hipStream_t stream) {
    const float* x      = (const float*)d_in[0];
    const int*   ei     = (const int*)d_in[1];   // [2, E] row-major (int32 per harness)
    const float* encW   = (const float*)d_in[2];
    const float* encB   = (const float*)d_in[3];
    const float* layerW = (const float*)d_in[4]; // [4, 256, 128]
    const float* layerB = (const float*)d_in[5]; // [4, 128]
    const float* decW1  = (const float*)d_in[6];
    const float* decB1  = (const float*)d_in[7];
    const float* decW2  = (const float*)d_in[8];
    const float* decB2  = (const float*)d_in[9];
    float* out = (float*)d_out;

    int n = in_sizes[0] / 16;
    int E = in_sizes[1] / 2;
    const int* rowi = ei;
    const int* coli = ei + E;

    // workspace: h [N*128] | agg [N*128] | deg [N]
    float* h   = (float*)d_ws;
    float* agg = h + (size_t)n * 128;
    float* deg = agg + (size_t)n * 128;

    int tiles = (n + 127) / 128;               // 8 waves/block x 16 rows/wave
    long featTot = (long)n * 128;

    zero_kernel<<<1024, 256, 0, stream>>>(deg, (long)n / 4);
    encoder_kernel<<<tiles, 256, 0, stream>>>(x, encW, encB, h, n);
    degree_kernel<<<(E + 255) / 256, 256, 0, stream>>>(coli, deg, E);
    invdeg_kernel<<<(n + 255) / 256, 256, 0, stream>>>(deg, n);

    long scatTot = (long)E * 32;
    int scatBlocks = (int)((scatTot + 255) / 256);
    if (scatBlocks > (1 << 20)) scatBlocks = 1 << 20;

    for (int l = 0; l < 4; ++l) {
        zero_kernel<<<4096, 256, 0, stream>>>(agg, featTot / 4);
        scatter_kernel<<<scatBlocks, 256, 0, stream>>>(h, agg, rowi, coli, scatTot);
        tile_gemm_kernel<8, true><<<tiles, 256, 0, stream>>>(
            h, agg, deg, layerW + (size_t)l * 256 * 128, layerB + (size_t)l * 128, h, n);
    }

    // decoder: g1 = gelu(h @ W1 + b1) reuses the agg buffer, then [128->4]
    tile_gemm_kernel<4, false><<<tiles, 256, 0, stream>>>(
        h, nullptr, nullptr, decW1, decB1, agg, n);
    dec2_kernel<<<(n + 255) / 256, 256, 0, stream>>>(agg, decW2, decB2, out, n);
}